// ParaPotentialNeuralNet_4320737100385
// MI455X (gfx1250) — compile-verified
//
#include <hip/hip_runtime.h>

typedef float v2f __attribute__((ext_vector_type(2)));
typedef float v8f __attribute__((ext_vector_type(8)));

constexpr int kNTA = 100000;   // atoms
constexpr int kNC  = 1000;     // crystals
constexpr int kNO  = 128;      // descriptor width (K of GEMM1)
constexpr int kH   = 166;      // hidden width
constexpr int kHP  = 176;      // hidden padded to multiple of 16
constexpr int kS   = 4;        // species
constexpr int kM   = 32;       // atoms per block (two 16-row M-tiles)

// One block == 32 atoms (two WMMA M-tiles). 128 threads == 4 wave32.
// Each wave loads one B fragment and feeds it to TWO wmma ops (one per M-tile),
// halving weight traffic per atom vs a 16-atom tile.
__global__ __launch_bounds__(128)
void nnp_fused(const float* __restrict__ positions,
               const int*   __restrict__ symbols,
               const float* __restrict__ Wd,     // [3, kNO]
               const float* __restrict__ W1,     // [kS, kNO, kH]
               const float* __restrict__ b1,     // [kS, kH]
               const float* __restrict__ W2,     // [kS, kH]
               const float* __restrict__ b2,     // [kS]
               float* __restrict__ out_energies, // [kNTA]
               float* __restrict__ out_forces)   // [kNTA, 3]
{
  // Region A: desc (GEMM1 A) -> reused as ddesc after last GEMM1 (barrier-separated)
  __shared__ __align__(16) float s_bufA[kM * kNO];     // 16 KB
  // Region B: pre1 (GEMM1 out) -> reused as energy/force partials after last activation
  __shared__ __align__(16) float s_bufB[kM * kHP];     // 22.5 KB
  __shared__ __align__(16) float s_g[kM][kHP];         // masked gate (GEMM2 A), zero padded
  __shared__ __align__(16) float s_wd[3 * kNO];
  __shared__ float s_pos[kM][3];
  __shared__ int   s_sym[kM];

  float (*s_desc)[kNO] = (float (*)[kNO])s_bufA;
  float (*s_dd)  [kNO] = (float (*)[kNO])s_bufA;            // alias (after barrier)
  float (*s_pre1)[kHP] = (float (*)[kHP])s_bufB;
  float (*s_ep)  [4]    = (float (*)[4])s_bufB;             // alias: [kM][4]
  float (*s_fp)  [4][3] = (float (*)[4][3])(s_bufB + kM*4); // alias: [kM][4][3]

  const int tid  = threadIdx.x;
  const int wave = tid >> 5;
  const int lane = tid & 31;
  const int base = blockIdx.x * kM;

  // ---- stage small inputs ----
  for (int i = tid; i < 3*kNO; i += 128) s_wd[i] = Wd[i];
  if (tid < kM) {
    int a = base + tid;
    bool val = a < kNTA;
    s_sym[tid]    = val ? symbols[a]       : 0;
    s_pos[tid][0] = val ? positions[a*3+0] : 0.f;
    s_pos[tid][1] = val ? positions[a*3+1] : 0.f;
    s_pos[tid][2] = val ? positions[a*3+2] : 0.f;
  }
  for (int i = tid; i < kM*kHP; i += 128) ((float*)s_g)[i] = 0.f; // incl. pad cols
  __syncthreads();

  // ---- descriptor: desc = sin(pos @ Wd) ----
  {
    int r = tid >> 2, oc = tid & 3;          // 32 rows x 4 col-groups
    float px = s_pos[r][0], py = s_pos[r][1], pz = s_pos[r][2];
    #pragma unroll 8
    for (int j = 0; j < 32; ++j) {
      int o = oc*32 + j;
      float p = px*s_wd[o] + py*s_wd[kNO+o] + pz*s_wd[2*kNO+o];
      s_desc[r][o] = __sinf(p);
    }
  }
  __syncthreads();

  // WMMA fp32 16x16x4 lane roles
  const int m16   = lane & 15;          // A row (within tile) / B,C,D column
  const int koff  = (lane >> 4) * 2;    // K base for this lane half
  const int mbase = (lane >> 4) * 8;    // C/D row base for this lane half

  v8f zero8 = {};
  v8f cdd[4]; cdd[0] = zero8; cdd[1] = zero8; cdd[2] = zero8; cdd[3] = zero8;

  float ep = 0.f;                 // energy partial for (row er, col group ec)
  const int er = tid & 31;        // activation row 0..31
  const int ec = tid >> 5;        // activation col group 0..3
  const int mysym = s_sym[er];

  for (int s = 0; s < kS; ++s) {
    const float* W1s = W1 + (size_t)s * kNO * kH;
    __builtin_prefetch(W1s, 0, 1);   // -> global_prefetch_b8

    // ---- GEMM1: pre1[32, kH] = desc[32,128] @ W1s[128, kH] ----
    for (int nt = wave; nt < (kHP/16); nt += 4) {
      int  n  = nt*16 + m16;
      bool nv = n < kH;
      int  nc = nv ? n : (kH - 1);                   // clamp: unconditional loads
      const float* Bp = W1s + (size_t)koff*kH + nc;
      v8f acc0 = zero8, acc1 = zero8;
      #pragma unroll 8
      for (int k0 = 0; k0 < kNO; k0 += 4) {
        v2f a0 = *(const v2f*)&s_desc[m16     ][k0 + koff];
        v2f a1 = *(const v2f*)&s_desc[m16 + 16][k0 + koff];
        v2f b; b.x = Bp[0]; b.y = Bp[kH];            // rows k, k+1 of column nc
        Bp += (size_t)4*kH;
        acc0 = __builtin_amdgcn_wmma_f32_16x16x4_f32(false, a0, false, b,
                                                     (short)0, acc0, false, false);
        acc1 = __builtin_amdgcn_wmma_f32_16x16x4_f32(false, a1, false, b,
                                                     (short)0, acc1, false, false);
      }
      if (nv) {
        #pragma unroll
        for (int v = 0; v < 8; ++v) {
          s_pre1[mbase + v     ][n] = acc0[v];
          s_pre1[mbase + v + 16][n] = acc1[v];
        }
      }
    }
    __syncthreads();

    // ---- activation: energy partials + masked backward gate g ----
    {
      const float* b1s = b1 + s*kH;
      const float* W2s = W2 + s*kH;
      bool match = (mysym == s);
      for (int col = ec; col < kH; col += 4) {
        float z   = s_pre1[er][col] + b1s[col];
        float sig = 1.f / (1.f + __expf(-z));
        float hh  = z * sig;                               // silu
        float w2v = W2s[col];
        float g   = w2v * sig * (1.f + z * (1.f - sig));   // w2 * silu'
        if (match) ep += hh * w2v;
        s_g[er][col] = match ? g : 0.f;
      }
    }
    __syncthreads();

    // ---- GEMM2: ddesc[32,128] += g[32,kHP] @ W1s^T[kHP,128] ----
    #pragma unroll
    for (int i = 0; i < 2; ++i) {
      int n = (wave + 4*i)*16 + m16;               // n < 128 always
      const float* W1Tn = W1s + (size_t)n * kH;    // W1^T(k,n) = W1s[n*kH + k]
      v8f acc0 = cdd[2*i], acc1 = cdd[2*i + 1];
      #pragma unroll 4
      for (int k0 = 0; k0 < kHP; k0 += 4) {
        int k  = k0 + koff;
        int kc = (k < kH - 1) ? k : (kH - 2);      // clamp; A is zero past kH
        v2f b  = *(const v2f*)(W1Tn + kc);         // contiguous pair: b64 load
        v2f a0 = *(const v2f*)&s_g[m16     ][k];
        v2f a1 = *(const v2f*)&s_g[m16 + 16][k];
        acc0 = __builtin_amdgcn_wmma_f32_16x16x4_f32(false, a0, false, b,
                                                     (short)0, acc0, false, false);
        acc1 = __builtin_amdgcn_wmma_f32_16x16x4_f32(false, a1, false, b,
                                                     (short)0, acc1, false, false);
      }
      cdd[2*i] = acc0; cdd[2*i + 1] = acc1;
    }
  }

  // ---- spill ddesc fragments (region A now free) and energy partials (region B) ----
  #pragma unroll
  for (int i = 0; i < 2; ++i) {
    int n = (wave + 4*i)*16 + m16;
    #pragma unroll
    for (int v = 0; v < 8; ++v) {
      s_dd[mbase + v     ][n] = cdd[2*i    ][v];
      s_dd[mbase + v + 16][n] = cdd[2*i + 1][v];
    }
  }
  s_ep[er][ec] = ep;
  __syncthreads();

  // ---- forces: dpre = ddesc * cos(pos@Wd); f = dpre @ Wd^T ----
  {
    int r = tid >> 2, oc = tid & 3;
    float px = s_pos[r][0], py = s_pos[r][1], pz = s_pos[r][2];
    float fx = 0.f, fy = 0.f, fz = 0.f;
    #pragma unroll 8
    for (int j = 0; j < 32; ++j) {
      int o = oc*32 + j;
      float w0 = s_wd[o], w1 = s_wd[kNO+o], w2 = s_wd[2*kNO+o];
      float p  = px*w0 + py*w1 + pz*w2;            // recompute pre (cheap)
      float dp = s_dd[r][o] * __cosf(p);
      fx += dp * w0; fy += dp * w1; fz += dp * w2;
    }
    s_fp[r][oc][0] = fx; s_fp[r][oc][1] = fy; s_fp[r][oc][2] = fz;
  }
  __syncthreads();

  if (tid < kM) {
    int a = base + tid;
    if (a < kNTA) {
      float e = 0.f;
      #pragma unroll
      for (int j = 0; j < 4; ++j) e += s_ep[tid][j];
      out_energies[a] = e + b2[s_sym[tid]];
    }
  }
  if (tid < 3*kM) {
    int r = tid / 3, c = tid % 3;
    int a = base + r;
    if (a < kNTA) {
      float f = 0.f;
      #pragma unroll
      for (int j = 0; j < 4; ++j) f += s_fp[r][j][c];
      out_forces[a*3 + c] = f;
    }
  }
}

// Deterministic segment-sum: one block per crystal (replay-stable, no float atomics).
__global__ __launch_bounds__(256)
void crystal_energy(const float* __restrict__ energies,
                    const int*   __restrict__ crystalidx,
                    float* __restrict__ out_energy)
{
  __shared__ float red[256];
  int c = blockIdx.x;
  float sum = 0.f;
  for (int i = threadIdx.x; i < kNTA; i += 256)
    sum += (crystalidx[i] == c) ? energies[i] : 0.f;
  red[threadIdx.x] = sum;
  __syncthreads();
  for (int off = 128; off > 0; off >>= 1) {
    if (threadIdx.x < off) red[threadIdx.x] += red[threadIdx.x + off];
    __syncthreads();
  }
  if (threadIdx.x == 0) out_energy[c] = red[0];
}

extern "C" void kernel_launch(void* const* d_in, const int* in_sizes, int n_in,
                              void* d_out, int out_size, void* d_ws, size_t ws_size,
                              hipStream_t stream)
{
  const float* positions  = (const float*)d_in[0];
  // d_in[1] = cells   (unused by the math)
  const int*   symbols    = (const int*)d_in[2];
  const int*   crystalidx = (const int*)d_in[3];
  // d_in[4] = pbcs    (unused by the math)
  const float* Wd = (const float*)d_in[5];
  const float* W1 = (const float*)d_in[6];
  const float* b1 = (const float*)d_in[7];
  const float* W2 = (const float*)d_in[8];
  const float* b2 = (const float*)d_in[9];

  float* out_energies = (float*)d_out;                 // [kNTA]
  float* out_energy   = out_energies + kNTA;           // [kNC]
  float* out_forces   = out_energy + kNC;              // [kNTA,3]

  int nblocks = (kNTA + kM - 1) / kM;
  nnp_fused<<<nblocks, 128, 0, stream>>>(positions, symbols, Wd, W1, b1, W2, b2,
                                         out_energies, out_forces);
  crystal_energy<<<kNC, 256, 0, stream>>>(out_energies, crystalidx, out_energy);
}